// PointerNetwork_5952824672534
// MI455X (gfx1250) — compile-verified
//
#include <hip/hip_runtime.h>
#include <math.h>

#define D      512
#define SPT    4
#define BSZ    32
#define SEQ    1024
#define SSUB   (SEQ*SPT)      // 4096
#define NSCORE (SSUB+1)       // 4097
#define VOCAB  32000
#define VEXT   34000
#define SCALE  0.04419417382415922f   // 1/sqrt(512)
#define EPSF   1.1920928955078125e-7f // float32 eps

typedef __attribute__((ext_vector_type(2))) float v2f;
typedef __attribute__((ext_vector_type(8))) float v8f;

// ---------------------------------------------------------------- utilities
__global__ void zero_kernel(float* __restrict__ p, int n) {
    int i = blockIdx.x * blockDim.x + threadIdx.x;
    if (i < n) p[i] = 0.0f;
}

// ---------------------------------------------------------------- q = tanh(query @ W_q + b_q)   (32 x 512)
__global__ __launch_bounds__(256) void q_kernel(const float* __restrict__ query,
                                                const float* __restrict__ W_q,
                                                const float* __restrict__ b_q,
                                                float* __restrict__ qout) {
    int idx = blockIdx.x * blockDim.x + threadIdx.x;   // 0..16383
    int b = idx >> 9, j = idx & (D - 1);
    const float* qr = query + b * D;
    const float* w  = W_q + j;                         // column j, coalesced across threads
    float acc = b_q[j];
    #pragma unroll 4
    for (int k = 0; k < D; ++k) acc += qr[k] * w[(size_t)k * D];
    qout[idx] = tanhf(acc);
}

// ---------------------------------------------------------------- v[b,k,t] = sum_dd W_sub[k, dd*4+t] * q[b,dd]
__global__ __launch_bounds__(256) void v_kernel(const float* __restrict__ W_sub,
                                                const float* __restrict__ qv,
                                                float* __restrict__ v) {
    int idx = blockIdx.x * blockDim.x + threadIdx.x;   // 0..65535 = b*2048 + k*4 + t
    int t = idx & 3, k = (idx >> 2) & (D - 1), b = idx >> 11;
    const float* w  = W_sub + (size_t)k * (SPT * D) + t;
    const float* qb = qv + b * D;
    float acc = 0.0f;
    #pragma unroll 4
    for (int dd = 0; dd < D; ++dd) acc += w[dd * 4] * qb[dd];
    v[idx] = acc;                                      // layout [b][k][t]
}

// ---------------------------------------------------------------- cbias[b,t] = sum_dd b_sub[dd*4+t] * q[b,dd]
__global__ void cbias_kernel(const float* __restrict__ b_sub,
                             const float* __restrict__ qv,
                             float* __restrict__ cb) {
    int idx = threadIdx.x;                             // 0..127 = b*4 + t
    if (idx >= BSZ * SPT) return;
    int t = idx & 3, b = idx >> 2;
    const float* qb = qv + b * D;
    float acc = 0.0f;
    for (int dd = 0; dd < D; ++dd) acc += b_sub[dd * 4 + t] * qb[dd];
    cb[idx] = acc;
}

// ---------------------------------------------------------------- scores via fp32 WMMA:  X_b(32x512) x Vpad_b(512x16)
// one wave per (batch, 32-row tile): two independent 16x16x4 accumulator
// chains (ILP, hides XDL latency). B operand loads are branchless (clamped
// column + select) so the wave stays convergent through the hot loop.
__global__ __launch_bounds__(32) void scores_wmma_kernel(const float* __restrict__ x,
                                                         const float* __restrict__ v,
                                                         const float* __restrict__ cb,
                                                         float* __restrict__ scores) {
    const int b    = blockIdx.y;
    const int s0   = blockIdx.x * 32;
    const int lane = threadIdx.x;          // 0..31
    const int half = lane >> 4;            // 0|1
    const int l    = lane & 15;
    const int lc   = l & 3;                // clamped B column
    const bool valid = (l < SPT);

    // this lane's x rows for the two 16-row sub-tiles
    const float* xr0 = x + ((size_t)b * SEQ + s0 + l) * D;
    const float* xr1 = xr0 + 16 * D;
    const float* vb  = v + (size_t)b * D * SPT;

    v8f acc0 = {0.f, 0.f, 0.f, 0.f, 0.f, 0.f, 0.f, 0.f};
    v8f acc1 = {0.f, 0.f, 0.f, 0.f, 0.f, 0.f, 0.f, 0.f};

    for (int k0 = 0; k0 < D; k0 += 4) {
        const int kk = k0 + 2 * half;      // lanes 0-15 -> K={k0,k0+1}; 16-31 -> K={k0+2,k0+3}
        // A 16x4 operands (aligned 8-byte loads, each x element read once)
        v2f a0 = *(const v2f*)(xr0 + kk);
        v2f a1 = *(const v2f*)(xr1 + kk);
        // B 4x16 operand: rows {kk, kk+1}, col l; cols >= 4 are zero pad.
        // Unconditional loads from clamped address, then select -> no exec divergence.
        float bl0 = vb[kk * SPT + lc];
        float bl1 = vb[(kk + 1) * SPT + lc];
        v2f bm;
        bm.x = valid ? bl0 : 0.0f;
        bm.y = valid ? bl1 : 0.0f;
        acc0 = __builtin_amdgcn_wmma_f32_16x16x4_f32(false, a0, false, bm,
                                                     (short)0, acc0, false, false);
        acc1 = __builtin_amdgcn_wmma_f32_16x16x4_f32(false, a1, false, bm,
                                                     (short)0, acc1, false, false);
    }

    // C/D: VGPR r -> (M = r + 8*half, N = l); we need N = 0..3 (the sub-token index)
    if (valid) {
        const float bias = cb[b * SPT + l];
        float* sc = scores + (size_t)b * NSCORE;
        #pragma unroll
        for (int r = 0; r < 8; ++r) {
            int m = r + 8 * half;
            sc[(s0 + m) * SPT + l]      = (acc0[r] + bias) * SCALE;
            sc[(s0 + 16 + m) * SPT + l] = (acc1[r] + bias) * SCALE;
        }
    }
}

// ---------------------------------------------------------------- per-batch: sentinel score, log-softmax over 4097,
// gate, logits logsumexp, scatter exp(dist) into p
__global__ __launch_bounds__(1024) void prep_kernel(const float* __restrict__ sentinel,
                                                    const float* __restrict__ qv,
                                                    const float* __restrict__ logits,
                                                    const int* __restrict__ evids,
                                                    float* __restrict__ scores,
                                                    float* __restrict__ p,
                                                    float* __restrict__ gate_arr,
                                                    float* __restrict__ slog_arr) {
    const int b = blockIdx.x, tid = threadIdx.x;
    __shared__ float red[1024];
    __shared__ float sval;

    float* sc = scores + (size_t)b * NSCORE;
    const float* qb = qv + b * D;

    // 1) sentinel score = dot(sentinel, q[b]) / sqrt(d)
    float acc = (tid < D) ? sentinel[tid] * qb[tid] : 0.0f;
    red[tid] = acc; __syncthreads();
    for (int off = 512; off > 0; off >>= 1) {
        if (tid < off) red[tid] += red[tid + off];
        __syncthreads();
    }
    if (tid == 0) { sval = red[0] * SCALE; sc[SSUB] = sval; }
    __syncthreads();
    const float ssent = sval;
    __syncthreads();

    // 2) max over 4097 scores
    float m = ssent;
    for (int i = tid; i < SSUB; i += 1024) m = fmaxf(m, sc[i]);
    red[tid] = m; __syncthreads();
    for (int off = 512; off > 0; off >>= 1) {
        if (tid < off) red[tid] = fmaxf(red[tid], red[tid + off]);
        __syncthreads();
    }
    if (tid == 0) sval = red[0];
    __syncthreads();
    m = sval;
    __syncthreads();

    // 3) logZ = m + log(sum exp(s - m))
    float s = (tid == 0) ? expf(ssent - m) : 0.0f;
    for (int i = tid; i < SSUB; i += 1024) s += expf(sc[i] - m);
    red[tid] = s; __syncthreads();
    for (int off = 512; off > 0; off >>= 1) {
        if (tid < off) red[tid] += red[tid + off];
        __syncthreads();
    }
    if (tid == 0) sval = m + logf(red[0]);
    __syncthreads();
    const float logZ = sval;
    __syncthreads();
    if (tid == 0) gate_arr[b] = ssent - logZ;

    // 4) logsumexp of subtoken_logits[b]
    const float* lg = logits + (size_t)b * VOCAB;
    float m2 = -__builtin_inff();
    for (int i = tid; i < VOCAB; i += 1024) m2 = fmaxf(m2, lg[i]);
    red[tid] = m2; __syncthreads();
    for (int off = 512; off > 0; off >>= 1) {
        if (tid < off) red[tid] = fmaxf(red[tid], red[tid + off]);
        __syncthreads();
    }
    if (tid == 0) sval = red[0];
    __syncthreads();
    m2 = sval;
    __syncthreads();
    float s2 = 0.0f;
    for (int i = tid; i < VOCAB; i += 1024) s2 += expf(lg[i] - m2);
    red[tid] = s2; __syncthreads();
    for (int off = 512; off > 0; off >>= 1) {
        if (tid < off) red[tid] += red[tid + off];
        __syncthreads();
    }
    if (tid == 0) slog_arr[b] = m2 + logf(red[0]);

    // 5) scatter exp(dist) into p[b, evids[b, i]]
    const int* ev = evids + (size_t)b * SSUB;
    float* pb = p + (size_t)b * VEXT;
    for (int i = tid; i < SSUB; i += 1024)
        atomicAdd(pb + ev[i], expf(sc[i] - logZ));
}

// ---------------------------------------------------------------- out[b,j] = logaddexp(lp+gate, log(p+eps))
__global__ __launch_bounds__(256) void final_kernel(const float* __restrict__ logits,
                                                    const float* __restrict__ p,
                                                    const float* __restrict__ gate_arr,
                                                    const float* __restrict__ slog_arr,
                                                    float* __restrict__ out) {
    int idx = blockIdx.x * blockDim.x + threadIdx.x;
    if (idx >= BSZ * VEXT) return;
    int b = idx / VEXT, j = idx - b * VEXT;
    float c = logf(p[idx] + EPSF);           // == pa + log(1-exp(gate)+eps), always finite
    float r;
    if (j < VOCAB) {
        float a  = logits[(size_t)b * VOCAB + j] - slog_arr[b] + gate_arr[b];
        float mx = fmaxf(a, c), mn = fminf(a, c);
        r = mx + log1pf(expf(mn - mx));
    } else {
        r = c;                               // lp = -inf branch
    }
    out[idx] = r;
}

// ---------------------------------------------------------------- launcher
extern "C" void kernel_launch(void* const* d_in, const int* in_sizes, int n_in,
                              void* d_out, int out_size, void* d_ws, size_t ws_size,
                              hipStream_t stream) {
    const float* x        = (const float*)d_in[0];
    const float* query    = (const float*)d_in[1];
    const float* logits   = (const float*)d_in[2];
    const int*   evids    = (const int*)  d_in[3];
    const float* W_sub    = (const float*)d_in[6];
    const float* b_sub    = (const float*)d_in[7];
    const float* W_q      = (const float*)d_in[8];
    const float* b_q      = (const float*)d_in[9];
    const float* sentinel = (const float*)d_in[10];
    float* out = (float*)d_out;

    float* ws     = (float*)d_ws;
    float* q_buf  = ws;                        // 32*512
    float* v_buf  = q_buf  + BSZ * D;          // 32*512*4
    float* cb_buf = v_buf  + BSZ * D * SPT;    // 32*4
    float* sc_buf = cb_buf + BSZ * SPT;        // 32*4097
    float* gate_b = sc_buf + BSZ * NSCORE;     // 32
    float* slog_b = gate_b + BSZ;              // 32
    float* p_buf  = slog_b + BSZ;              // 32*34000   (total ~5.2 MB)

    zero_kernel<<<(BSZ * VEXT + 255) / 256, 256, 0, stream>>>(p_buf, BSZ * VEXT);
    q_kernel<<<(BSZ * D) / 256, 256, 0, stream>>>(query, W_q, b_q, q_buf);
    cbias_kernel<<<1, 128, 0, stream>>>(b_sub, q_buf, cb_buf);
    v_kernel<<<(BSZ * D * SPT) / 256, 256, 0, stream>>>(W_sub, q_buf, v_buf);
    scores_wmma_kernel<<<dim3(SEQ / 32, BSZ), 32, 0, stream>>>(x, v_buf, cb_buf, sc_buf);
    prep_kernel<<<BSZ, 1024, 0, stream>>>(sentinel, q_buf, logits, evids,
                                          sc_buf, p_buf, gate_b, slog_b);
    final_kernel<<<(BSZ * VEXT + 255) / 256, 256, 0, stream>>>(logits, p_buf, gate_b,
                                                               slog_b, out);
}